// HadaMard_46866683134193
// MI455X (gfx1250) — compile-verified
//
#include <hip/hip_runtime.h>

typedef __attribute__((ext_vector_type(2))) float v2f;
typedef __attribute__((ext_vector_type(4))) float f4;
typedef __attribute__((ext_vector_type(8))) float v8f;

#define B_   8
#define C1_  1024
#define C2_  32
#define CP_  256
#define N_   1024
#define EPS_ 1e-5f

// ---------------------------------------------------------------------------
// Kernel 1: tmp[b,d,n] = sum_c Wp[d,c] * q[b,c,n] + bp[d]
// One wave per 16x16 (d,n) tile; K swept 4 at a time with V_WMMA_F32_16X16X4_F32.
// 8192 tiles total = 1024 blocks x 8 waves.
// ---------------------------------------------------------------------------
__global__ __launch_bounds__(256) void proj_wmma_kernel(
    const float* __restrict__ q, const float* __restrict__ Wp,
    const float* __restrict__ bp, float* __restrict__ tmp) {
  const int lane = threadIdx.x & 31;
  const int wave = threadIdx.x >> 5;
  const int gw   = blockIdx.x * 8 + wave;      // 0..8191
  const int b    = gw >> 10;                   // 0..7
  const int rem  = gw & 1023;
  const int dT   = rem >> 6;                   // 0..15
  const int nT   = rem & 63;                   // 0..63
  const int half = lane >> 4;                  // 0 or 1
  const int l16  = lane & 15;
  const int dRow = dT * 16;
  const int ncol = nT * 16 + l16;

  const float* qB   = q + (size_t)b * C1_ * N_;
  // A fragment source: this lane's Wp row, offset by its K half (K = 2*half + {0,1})
  const float* wRow = Wp + (size_t)(dRow + l16) * C1_ + 2 * half;
  // B fragment source: this lane's q column, offset by its K half
  const float* qCol = qB + ncol + (size_t)(2 * half) * N_;

  v8f acc[4];
#pragma unroll
  for (int j = 0; j < 4; ++j) acc[j] = {};

#pragma unroll 2
  for (int k = 0; k < C1_; k += 16) {
#pragma unroll
    for (int j = 0; j < 4; ++j) {
      const int kk = k + 4 * j;
      v2f a = *(const v2f*)(wRow + kk);        // 8B aligned (kk even, 4KB rows)
      v2f bb;
      bb.x = qCol[(size_t)kk * N_];
      bb.y = qCol[(size_t)(kk + 1) * N_];
      acc[j] = __builtin_amdgcn_wmma_f32_16x16x4_f32(
          false, a, false, bb, (short)0, acc[j], false, false);
    }
  }

  v8f s = acc[0] + acc[1] + acc[2] + acc[3];

  float* o = tmp + ((size_t)b * CP_ + dRow) * N_ + ncol;
#pragma unroll
  for (int v = 0; v < 8; ++v) {
    const int d = v + 8 * half;                // C/D layout: M = v + 8*half, N = l16
    o[(size_t)d * N_] = s[v] + bp[dRow + d];
  }
}

// ---------------------------------------------------------------------------
// Kernel 2: in-place LayerNorm of tmp over the d (=Cp) axis per (b,n) column.
// Thread per (b,n); loads coalesced across consecutive-n threads.
// ---------------------------------------------------------------------------
__global__ __launch_bounds__(256) void norm_q_kernel(
    float* __restrict__ tmp, const float* __restrict__ g1,
    const float* __restrict__ b1) {
  const int tid = blockIdx.x * blockDim.x + threadIdx.x;  // 0..8191
  const int b = tid >> 10, n = tid & 1023;
  float* col = tmp + (size_t)b * CP_ * N_ + n;
  float s = 0.f, ss = 0.f;
#pragma unroll 8
  for (int d = 0; d < CP_; ++d) {
    float v = col[(size_t)d * N_];
    s += v;
    ss = fmaf(v, v, ss);
  }
  const float m   = s * (1.0f / CP_);
  const float var = ss * (1.0f / CP_) - m * m;
  const float r   = rsqrtf(var + EPS_);
#pragma unroll 8
  for (int d = 0; d < CP_; ++d) {
    float v = col[(size_t)d * N_];
    col[(size_t)d * N_] = (v - m) * r * g1[d] + b1[d];
  }
}

// ---------------------------------------------------------------------------
// Kernel 3: xn[b,c,n] = LayerNorm over c of x[b,c,n].
// ---------------------------------------------------------------------------
__global__ __launch_bounds__(256) void norm_x_kernel(
    const float* __restrict__ x, float* __restrict__ xn,
    const float* __restrict__ g2, const float* __restrict__ b2) {
  const int tid = blockIdx.x * blockDim.x + threadIdx.x;  // 0..8191
  const int b = tid >> 10, n = tid & 1023;
  const float* col = x + (size_t)b * C2_ * N_ + n;
  float* oc = xn + (size_t)b * C2_ * N_ + n;
  float s = 0.f, ss = 0.f;
#pragma unroll
  for (int c = 0; c < C2_; ++c) {
    float v = col[(size_t)c * N_];
    s += v;
    ss = fmaf(v, v, ss);
  }
  const float m   = s * (1.0f / C2_);
  const float var = ss * (1.0f / C2_) - m * m;
  const float r   = rsqrtf(var + EPS_);
#pragma unroll
  for (int c = 0; c < C2_; ++c) {
    float v = col[(size_t)c * N_];
    oc[(size_t)c * N_] = (v - m) * r * g2[c] + b2[c];
  }
}

// ---------------------------------------------------------------------------
// Kernel 4 (the 256 MB write): out[r, n] = qn[b,cp,n] * xn[b,c2,n]
// r = (b*Cp + cp)*C2 + c2. One block per output row, float4 per thread,
// nontemporal stores (output never re-read; keep L2 for qn/xn reuse).
// ---------------------------------------------------------------------------
__global__ __launch_bounds__(256) void outer_kernel(
    const float* __restrict__ qn, const float* __restrict__ xn,
    float* __restrict__ out) {
  const int r   = blockIdx.x;                  // 0..65535
  const int b   = r >> 13;                     // / 8192
  const int rem = r & 8191;
  const int cp  = rem >> 5;
  const int c2  = rem & 31;
  const f4* qrow = (const f4*)(qn + ((size_t)b * CP_ + cp) * N_);
  const f4* xrow = (const f4*)(xn + ((size_t)b * C2_ + c2) * N_);
  const f4 a = qrow[threadIdx.x];
  const f4 c = xrow[threadIdx.x];
  const f4 o = a * c;
  f4* orow = (f4*)(out + (size_t)r * N_);
  __builtin_nontemporal_store(o, orow + threadIdx.x);
}

// ---------------------------------------------------------------------------
extern "C" void kernel_launch(void* const* d_in, const int* in_sizes, int n_in,
                              void* d_out, int out_size, void* d_ws,
                              size_t ws_size, hipStream_t stream) {
  const float* q  = (const float*)d_in[0];  // [8,1024,32,32]
  const float* x  = (const float*)d_in[1];  // [8,32,32,32]
  const float* Wp = (const float*)d_in[2];  // [256,1024]
  const float* bp = (const float*)d_in[3];  // [256]
  const float* g1 = (const float*)d_in[4];  // [256]
  const float* b1 = (const float*)d_in[5];  // [256]
  const float* g2 = (const float*)d_in[6];  // [32]
  const float* b2 = (const float*)d_in[7];  // [32]
  float* out = (float*)d_out;

  float* tmp = (float*)d_ws;                          // 8*256*1024 = 8 MiB
  float* xn  = tmp + (size_t)B_ * CP_ * N_;           // 8*32*1024  = 1 MiB

  // 1) projection GEMM via WMMA: 8192 tiles, 8 waves/block
  proj_wmma_kernel<<<dim3(1024), dim3(256), 0, stream>>>(q, Wp, bp, tmp);
  // 2) LayerNorm over Cp, in place
  norm_q_kernel<<<dim3(32), dim3(256), 0, stream>>>(tmp, g1, b1);
  // 3) LayerNorm of x over C2
  norm_x_kernel<<<dim3(32), dim3(256), 0, stream>>>(x, xn, g2, b2);
  // 4) outer product + 256 MB streaming write
  outer_kernel<<<dim3(B_ * CP_ * C2_), dim3(256), 0, stream>>>(tmp, xn, out);
}